// GeometricAttention_83708912599263
// MI455X (gfx1250) — compile-verified
//
#include <hip/hip_runtime.h>
#include <hip/hip_bf16.h>

typedef __attribute__((ext_vector_type(2))) float v2f;
typedef __attribute__((ext_vector_type(8))) float v8f;

#define NB 2
#define LSEQ 768
#define DDIM 128
#define PDIM 64
#define HH 12
#define QKD 16
#define VDD 16
#define HQK 192            // H*QK = H*VD
#define FEAT 1044          // H*P + H*VD + H*7
#define INF_ 100000.0f
#define SPATIAL_SCALE 0.23570226039551584f  // sqrt(2/9)/2
#define LOGIT_SCALE  0.5773502691896258f    // sqrt(1/3)

#define AS3 __attribute__((address_space(3)))

// LDS byte-offset of a generic pointer into shared memory (VDST operand of
// the async global->LDS instructions is an LDS byte address).
static __device__ __forceinline__ unsigned ldsOff(const void* p) {
    return (unsigned)(unsigned long long)(AS3 const void*)p;
}

// global_load_async_to_lds_b128: per-lane 16B memory -> LDS, ASYNCcnt-tracked.
static __device__ __forceinline__ void asyncCopy16(const float* g, unsigned lds_byte) {
    asm volatile("global_load_async_to_lds_b128 %0, %1, off"
                 :: "v"(lds_byte), "v"(g)
                 : "memory");
}
static __device__ __forceinline__ void waitAsync8() {
    asm volatile("s_wait_asynccnt 0x8" ::: "memory");
}
static __device__ __forceinline__ void waitAsync0() {
    asm volatile("s_wait_asynccnt 0x0" ::: "memory");
}
static __device__ __forceinline__ void waitDs0() {
    asm volatile("s_wait_dscnt 0x0" ::: "memory");
}

// ---------------------------------------------------------------------------
// Kernel 1: q/k/v = x @ {Wq,Wk,Wv}.  One 16x16 tile per wave via chained
// V_WMMA_F32_16X16X4_F32 over K=128 (32 WMMA per tile, fp32 end-to-end).
// fp32 A-frag layout (ISA 7.12.2): lanes 0-15 hold K=+0,+1; lanes 16-31 K=+2,+3.
// ---------------------------------------------------------------------------
__global__ __launch_bounds__(128)
void qkv_proj_kernel(const float* __restrict__ x,
                     const float* __restrict__ Wq,
                     const float* __restrict__ Wk,
                     const float* __restrict__ Wv,
                     float* __restrict__ qws,
                     float* __restrict__ kws,
                     float* __restrict__ vws)
{
    const int lane = threadIdx.x & 31;
    const int wave = threadIdx.x >> 5;
    const int tile = blockIdx.x * 4 + wave;          // 0..3455
    const int ROWT = (NB * LSEQ) / 16;               // 96
    const int COLT = HQK / 16;                       // 12
    const int sel = tile / (ROWT * COLT);            // 0:q 1:k 2:v
    const int rem = tile - sel * (ROWT * COLT);
    const int rt = rem / COLT;
    const int ct = rem - rt * COLT;

    const float* __restrict__ W = (sel == 0) ? Wq : (sel == 1) ? Wk : Wv;
    float* __restrict__ O       = (sel == 0) ? qws : (sel == 1) ? kws : vws;

    const int r0 = rt * 16, c0 = ct * 16;
    const int mrow = r0 + (lane & 15);
    const int ncol = c0 + (lane & 15);
    const int koff = (lane < 16) ? 0 : 2;

    v8f acc = {};
    for (int kk = 0; kk < DDIM; kk += 4) {
        v2f a, b;
        a.x = x[mrow * DDIM + kk + koff];
        a.y = x[mrow * DDIM + kk + koff + 1];
        b.x = W[(kk + koff) * HQK + ncol];
        b.y = W[(kk + koff + 1) * HQK + ncol];
        acc = __builtin_amdgcn_wmma_f32_16x16x4_f32(false, a, false, b,
                                                    (short)0, acc, false, false);
    }
    #pragma unroll
    for (int j = 0; j < 8; ++j) {
        const int row = r0 + ((lane < 16) ? j : (8 + j));
        O[row * HQK + ncol] = acc[j];
    }
}

// ---------------------------------------------------------------------------
// Kernel 2: fused attention row.  One block per (b,l); 256 threads = 8 waves.
// Dynamic LDS (~108 KB, fine for CDNA5's 320KB WGP LDS):
//   lg[768*12]  logits->alpha panel    36 KB
//   zst         8 waves x 2 x (16x64)  64 KB   async-staged z tiles
//   featall/qrow/red/...               ~6 KB
// z streams from HBM exactly twice (logits pass via async->LDS staging,
// aggregation pass direct); nothing else round-trips memory.
// ---------------------------------------------------------------------------
__global__ __launch_bounds__(256)
void attn_fused_kernel(const float* __restrict__ x,
                       const float* __restrict__ z,
                       const unsigned char* __restrict__ maskp,
                       const float* __restrict__ pos_CB,
                       const float* __restrict__ pos_CA,
                       const float* __restrict__ frame,
                       const float* __restrict__ Wp,
                       const float* __restrict__ spatial_coef,
                       const float* __restrict__ Wo,
                       const float* __restrict__ bo,
                       const float* __restrict__ ln_g,
                       const float* __restrict__ ln_b,
                       const float* __restrict__ qws,
                       const float* __restrict__ kws,
                       const float* __restrict__ vws,
                       float* __restrict__ out)
{
    extern __shared__ float smem[];
    float* lg      = smem;                       //  9216 floats
    float* zst     = lg + LSEQ * HH;             // 16384 floats (8x2x1024)
    float* featall = zst + 16384;                //  1044
    float* qrow    = featall + FEAT;             //   192
    float* red     = qrow + HQK;                 //   256
    float* gammaS  = red + 256;                  //    12
    float* aggrS   = gammaS + HH;                //    36
    float* plCB    = aggrS + HH * 3;             //     3
    float* plCA    = plCB + 3;                   //     3
    float* frm     = plCA + 3;                   //     9

    const int tid  = threadIdx.x;
    const int lane = tid & 31;
    const int wave = tid >> 5;
    const int b = blockIdx.x / LSEQ;
    const int l = blockIdx.x - b * LSEQ;
    const int row = b * LSEQ + l;

    if (tid < HQK) qrow[tid] = qws[(size_t)row * HQK + tid];
    if (tid < HH)  gammaS[tid] = log1pf(__expf(spatial_coef[tid]));   // softplus
    if (tid >= 32 && tid < 35)  plCB[tid - 32] = pos_CB[(size_t)row * 3 + (tid - 32)];
    if (tid >= 64 && tid < 67)  plCA[tid - 64] = pos_CA[(size_t)row * 3 + (tid - 64)];
    if (tid >= 96 && tid < 105) frm[tid - 96]  = frame[(size_t)row * 9 + (tid - 96)];
    __syncthreads();

    const bool mrow_ok = maskp[row] != 0;
    const float* __restrict__ zrow = z + ((size_t)b * LSEQ + l) * (size_t)(LSEQ * PDIM);
    const int koff = (lane < 16) ? 0 : 2;
    const int hcol = lane & 15;

    // ---- Pass A: logits = (q.k + z@Wp + spatial)*scale, masked ------------
    // Each wave owns 6 k-tiles (16 keys x 64 pair-channels = 4KB each) and
    // double-buffers them LDS-side with global_load_async_to_lds_b128.
    {
        float* zb0 = zst + wave * 2048;
        float* zb1 = zb0 + 1024;
        const unsigned zo[2] = { ldsOff(zb0), ldsOff(zb1) };
        const int NT = (LSEQ / 16) / 8;   // 6 tiles per wave

        // prologue: stage tile 0 into buffer 0
        {
            const float* gsrc = zrow + (size_t)(wave) * 16 * PDIM;
            #pragma unroll
            for (int i = 0; i < 8; ++i)
                asyncCopy16(gsrc + (i * 32 + lane) * 4, zo[0] + (i * 32 + lane) * 16);
        }

        for (int t = 0; t < NT; ++t) {
            if (t + 1 < NT) {
                waitDs0();   // previous consume's ds_loads done before overwrite
                const float* gsrc = zrow + (size_t)(wave + 8 * (t + 1)) * 16 * PDIM;
                const unsigned dst = zo[(t + 1) & 1];
                #pragma unroll
                for (int i = 0; i < 8; ++i)
                    asyncCopy16(gsrc + (i * 32 + lane) * 4, dst + (i * 32 + lane) * 16);
                waitAsync8();   // tile t landed (8 newer issues still in flight)
            } else {
                waitAsync0();
            }

            const float* stg = (t & 1) ? zb1 : zb0;   // 16 rows x 64 cols in LDS
            const int k0 = (wave + 8 * t) * 16;

            v8f acc = {};                              // pair-logit tile (16k x 16h)
            for (int cc = 0; cc < PDIM; cc += 4) {
                v2f a, bf;
                a.x = stg[(lane & 15) * PDIM + cc + koff];        // ds_load
                a.y = stg[(lane & 15) * PDIM + cc + koff + 1];
                bf.x = (hcol < HH) ? Wp[(cc + koff) * HH + hcol] : 0.0f;
                bf.y = (hcol < HH) ? Wp[(cc + koff + 1) * HH + hcol] : 0.0f;
                acc = __builtin_amdgcn_wmma_f32_16x16x4_f32(false, a, false, bf,
                                                            (short)0, acc, false, false);
            }
            if (hcol < HH) {
                #pragma unroll
                for (int j = 0; j < 8; ++j) {
                    const int k = k0 + ((lane < 16) ? j : (8 + j));
                    const float* __restrict__ kv =
                        kws + (size_t)(b * LSEQ + k) * HQK + hcol * QKD;
                    float node = 0.0f;
                    #pragma unroll
                    for (int d0 = 0; d0 < QKD; ++d0)
                        node = fmaf(qrow[hcol * QKD + d0], kv[d0], node);
                    const float dx = plCB[0] - pos_CB[(size_t)(b * LSEQ + k) * 3 + 0];
                    const float dy = plCB[1] - pos_CB[(size_t)(b * LSEQ + k) * 3 + 1];
                    const float dz = plCB[2] - pos_CB[(size_t)(b * LSEQ + k) * 3 + 2];
                    const float d2 = dx * dx + dy * dy + dz * dz;
                    float lgt = (acc[j] + node - gammaS[hcol] * SPATIAL_SCALE * d2)
                                * LOGIT_SCALE;
                    const bool mk = mrow_ok && (maskp[b * LSEQ + k] != 0);
                    if (!mk) lgt -= INF_;
                    lg[k * HH + hcol] = lgt;
                }
            }
        }
    }
    __syncthreads();

    // ---- Softmax over keys, per head (in LDS) -----------------------------
    for (int h = 0; h < HH; ++h) {
        float m = -3.4e38f;
        for (int k = tid; k < LSEQ; k += 256) m = fmaxf(m, lg[k * HH + h]);
        red[tid] = m; __syncthreads();
        for (int s = 128; s > 0; s >>= 1) {
            if (tid < s) red[tid] = fmaxf(red[tid], red[tid + s]);
            __syncthreads();
        }
        const float M = red[0]; __syncthreads();
        float ssum = 0.0f;
        for (int k = tid; k < LSEQ; k += 256) {
            const float e = __expf(lg[k * HH + h] - M);
            lg[k * HH + h] = e;
            ssum += e;
        }
        red[tid] = ssum; __syncthreads();
        for (int s = 128; s > 0; s >>= 1) {
            if (tid < s) red[tid] += red[tid + s];
            __syncthreads();
        }
        const float inv = mrow_ok ? (1.0f / red[0]) : 0.0f;   // mask_row zeroing
        __syncthreads();
        for (int k = tid; k < LSEQ; k += 256) lg[k * HH + h] *= inv;
    }
    __syncthreads();

    // ---- Pass B: aggregate.  Waves 0-3: alpha^T @ z via WMMA chain. -------
    if (wave < 4) {
        const int c0 = wave * 16;
        const int ccol = c0 + (lane & 15);
        __builtin_prefetch(zrow + ccol, 0, 1);
        v8f acc = {};
        for (int kk = 0; kk < LSEQ; kk += 4) {
            v2f a, bf;
            a.x = (hcol < HH) ? lg[(kk + koff) * HH + hcol] : 0.0f;     // ds_load
            a.y = (hcol < HH) ? lg[(kk + koff + 1) * HH + hcol] : 0.0f;
            bf.x = zrow[(size_t)(kk + koff) * PDIM + ccol];
            bf.y = zrow[(size_t)(kk + koff + 1) * PDIM + ccol];
            acc = __builtin_amdgcn_wmma_f32_16x16x4_f32(false, a, false, bf,
                                                        (short)0, acc, false, false);
        }
        #pragma unroll
        for (int j = 0; j < 8; ++j) {
            const int h = (lane < 16) ? j : (8 + j);
            if (h < HH) featall[h * PDIM + ccol] = acc[j];              // feat_p2n
        }
    } else if (wave < 6) {            // waves 4-5: feat_node = alpha @ v (per head)
        const int t2 = tid - 128;     // 0..63
        for (int e = t2; e < HH * VDD; e += 64) {
            const int h = e / VDD, c = e - h * VDD;
            float acc = 0.0f;
            for (int k = 0; k < LSEQ; ++k)
                acc = fmaf(lg[k * HH + h],
                           vws[(size_t)(b * LSEQ + k) * HQK + h * VDD + c], acc);
            featall[HH * PDIM + e] = acc;
        }
    } else if (wave == 6) {           // wave 6: aggr = alpha @ pos_CB
        for (int e = lane; e < HH * 3; e += 32) {
            const int h = e / 3, d3 = e - h * 3;
            float acc = 0.0f;
            for (int k = 0; k < LSEQ; ++k)
                acc = fmaf(lg[k * HH + h],
                           pos_CB[(size_t)(b * LSEQ + k) * 3 + d3], acc);
            aggrS[e] = acc;
        }
    }
    __syncthreads();

    // ---- Spatial features: frame rotate, dist, direction ------------------
    if (tid < HH) {
        const int h = tid;
        const float px = aggrS[h * 3 + 0] - plCA[0];
        const float py = aggrS[h * 3 + 1] - plCA[1];
        const float pz = aggrS[h * 3 + 2] - plCA[2];
        const float f0 = frm[0] * px + frm[1] * py + frm[2] * pz;
        const float f1 = frm[3] * px + frm[4] * py + frm[5] * pz;
        const float f2 = frm[6] * px + frm[7] * py + frm[8] * pz;
        const float dist = sqrtf(f0 * f0 + f1 * f1 + f2 * f2);
        const float invd = 1.0f / (dist + 1.0e-4f);
        featall[960 + h * 3 + 0] = f0;
        featall[960 + h * 3 + 1] = f1;
        featall[960 + h * 3 + 2] = f2;
        featall[996 + h] = dist;
        featall[1008 + h * 3 + 0] = f0 * invd;
        featall[1008 + h * 3 + 1] = f1 * invd;
        featall[1008 + h * 3 + 2] = f2 * invd;
    }
    __syncthreads();

    // ---- Output GEMV (1044 -> 128) + residual + LayerNorm -----------------
    float hval = 0.0f;
    if (tid < DDIM) {
        float acc = bo[tid];
        for (int i = 0; i < FEAT; ++i)
            acc = fmaf(featall[i], Wo[(size_t)i * DDIM + tid], acc);
        if (!mrow_ok) acc = 0.0f;
        hval = x[(size_t)row * DDIM + tid] + acc;
    }
    red[tid] = (tid < DDIM) ? hval : 0.0f;
    __syncthreads();
    for (int s = 128; s > 0; s >>= 1) {
        if (tid < s) red[tid] += red[tid + s];
        __syncthreads();
    }
    const float mu = red[0] * (1.0f / DDIM);
    __syncthreads();
    const float dev = (tid < DDIM) ? (hval - mu) : 0.0f;
    red[tid] = dev * dev;
    __syncthreads();
    for (int s = 128; s > 0; s >>= 1) {
        if (tid < s) red[tid] += red[tid + s];
        __syncthreads();
    }
    const float var = red[0] * (1.0f / DDIM);
    if (tid < DDIM)
        out[(size_t)row * DDIM + tid] =
            (hval - mu) * rsqrtf(var + 1.0e-5f) * ln_g[tid] + ln_b[tid];
}

// ---------------------------------------------------------------------------
extern "C" void kernel_launch(void* const* d_in, const int* in_sizes, int n_in,
                              void* d_out, int out_size, void* d_ws, size_t ws_size,
                              hipStream_t stream) {
    const float* x      = (const float*)d_in[0];
    const float* z      = (const float*)d_in[1];
    const unsigned char* mask = (const unsigned char*)d_in[2];
    const float* pos_CB = (const float*)d_in[3];
    const float* pos_CA = (const float*)d_in[4];
    const float* frame  = (const float*)d_in[5];
    const float* Wq     = (const float*)d_in[6];
    const float* Wk     = (const float*)d_in[7];
    const float* Wv     = (const float*)d_in[8];
    const float* Wp     = (const float*)d_in[9];
    const float* sc     = (const float*)d_in[10];
    const float* Wo     = (const float*)d_in[11];
    const float* bo     = (const float*)d_in[12];
    const float* ln_g   = (const float*)d_in[13];
    const float* ln_b   = (const float*)d_in[14];
    float* out = (float*)d_out;

    float* qws = (float*)d_ws;
    float* kws = qws + (size_t)NB * LSEQ * HQK;
    float* vws = kws + (size_t)NB * LSEQ * HQK;

    // 3456 tiles (96 row-tiles x 12 col-tiles x {q,k,v}), 4 waves/block
    qkv_proj_kernel<<<864, 128, 0, stream>>>(x, Wq, Wk, Wv, qws, kws, vws);

    // dynamic LDS: lg 9216 + zst 16384 + featall 1044 + qrow 192 + red 256
    //            + gamma 12 + aggr 36 + plCB/plCA/frm 15  = 27155 floats
    const size_t shbytes = 27155u * sizeof(float);
    attn_fused_kernel<<<NB * LSEQ, 256, shbytes, stream>>>(
        x, z, mask, pos_CB, pos_CA, frame, Wp, sc, Wo, bo, ln_g, ln_b,
        qws, kws, vws, out);
}